// SinkhornKnopp_24137716203607
// MI455X (gfx1250) — compile-verified
//
#include <hip/hip_runtime.h>
#include <cstdint>
#include <cstddef>

// Sinkhorn-Knopp as diagonal scaling: M = diag(u) * exp(P0) * diag(v).
// Per iteration:  u = 1/(P0*v);  v = 1/(P0^T*u).  Tile is READ-ONLY in LDS.
// One 256x256 matrix per workgroup, fully resident in CDNA5's 320KB WGP LDS.
#define N 256
#define STRIDE 260                       // 256 + 4 pad: 16B-aligned rows, decent bank spread
#define PART_OFF (N * STRIDE)            // 66560: 16x256 partial-sum buffer (shared by both reduces)
#define U_OFF (PART_OFF + 16 * N)        // 70656: row scales
#define V_OFF (U_OFF + N)                // 70912: col scales
#define LDS_FLOATS (V_OFF + N)           // 71168
#define LDS_BYTES (LDS_FLOATS * 4)       // 284672 bytes < 320KB
#define NUM_ITERS 20

__global__ __launch_bounds__(1024, 1)
void sinkhorn_kernel(const float* __restrict__ in, float* __restrict__ out) {
    extern __shared__ float lds[];
    float* P    = lds;                   // exp(P0), 256 x 260, read-only after init
    float* part = lds + PART_OFF;        // partial sums (row phase uses 4 rows, col phase 16)
    float* u    = lds + U_OFF;
    float* v    = lds + V_OFF;

    const int t = threadIdx.x;           // 0..1023 (32 waves)
    const int g = blockIdx.x;
    const float* gin  = in  + (size_t)g * (N * N);
    float*       gout = out + (size_t)g * (N * N);

    const unsigned lds_base = (unsigned)(uintptr_t)P;   // low 32 bits = LDS byte offset

    // ---- Async HBM -> LDS fill: 16 x b128 per thread (ASYNCcnt-tracked DMA path) ----
#pragma unroll
    for (int k = 0; k < 16; ++k) {
        const int flat = k * 4096 + t * 4;              // row-major flat index
        const int r = flat >> 8;
        const int j = flat & 255;
        const unsigned lds_off = lds_base + (unsigned)((r * STRIDE + j) * 4);
        const unsigned voff    = (unsigned)(flat * 4);
        asm volatile("global_load_async_to_lds_b128 %0, %1, %2"
                     :: "v"(lds_off), "v"(voff), "s"(gin) : "memory");
    }

    // ---- exp() in place, pipelined against the DMA: async loads complete in
    // issue order, so asynccnt <= 15-k guarantees chunks 0..k have landed. ----
#pragma unroll
    for (int k = 0; k < 16; ++k) {
        asm volatile("s_wait_asynccnt %0" :: "i"(15 - k) : "memory");
        const int flat = k * 4096 + t * 4;
        const int r = flat >> 8;
        const int j = flat & 255;
        float4* p = (float4*)(P + r * STRIDE + j);
        float4 x = *p;
        x.x = __expf(x.x); x.y = __expf(x.y);
        x.z = __expf(x.z); x.w = __expf(x.w);
        *p = x;
    }
    if (t < N) v[t] = 1.0f;              // v starts as ones
    __syncthreads();

    // Row-matvec mapping: thread owns (row r_row, 64-wide segment seg_row)
    const int r_row = t & 255, seg_row = t >> 8;
    const float4* Prow = (const float4*)(P + r_row * STRIDE + seg_row * 64);
    const float4* vseg = (const float4*)(v + seg_row * 64);      // wave-broadcast reads

    // Col-matvec mapping: thread owns 4 adjacent columns c0..c0+3, 16 rows r0..r0+15
    const int c0 = (t & 63) * 4, seg_col = t >> 6;
    const int r0 = seg_col * 16;
    const float* Pcol = P + r0 * STRIDE + c0;

#pragma unroll 1
    for (int it = 0; it < NUM_ITERS; ++it) {
        // ---- u = 1 / (P0 * v): row matvec, float4 loads along rows ----
        {
            float4 a = make_float4(0.f, 0.f, 0.f, 0.f);
#pragma unroll
            for (int k = 0; k < 16; ++k) {
                float4 p = Prow[k];
                float4 w = vseg[k];
                a.x = fmaf(p.x, w.x, a.x);
                a.y = fmaf(p.y, w.y, a.y);
                a.z = fmaf(p.z, w.z, a.z);
                a.w = fmaf(p.w, w.w, a.w);
            }
            part[seg_row * 256 + r_row] = (a.x + a.y) + (a.z + a.w);
        }
        __syncthreads();
        if (t < N) {
            float s = part[t] + part[256 + t] + part[512 + t] + part[768 + t];
            u[t] = 1.0f / s;
        }
        __syncthreads();

        // ---- v = 1 / (P0^T * u): col matvec, float4 loads, 4 column-partials per lane ----
        {
            float4 u0 = *(const float4*)(u + r0);
            float4 u1 = *(const float4*)(u + r0 + 4);
            float4 u2 = *(const float4*)(u + r0 + 8);
            float4 u3 = *(const float4*)(u + r0 + 12);
            const float uu[16] = { u0.x, u0.y, u0.z, u0.w,  u1.x, u1.y, u1.z, u1.w,
                                   u2.x, u2.y, u2.z, u2.w,  u3.x, u3.y, u3.z, u3.w };
            float4 a = make_float4(0.f, 0.f, 0.f, 0.f);
#pragma unroll
            for (int i = 0; i < 16; ++i) {
                float4 p = *(const float4*)(Pcol + i * STRIDE);
                const float ui = uu[i];
                a.x = fmaf(p.x, ui, a.x);
                a.y = fmaf(p.y, ui, a.y);
                a.z = fmaf(p.z, ui, a.z);
                a.w = fmaf(p.w, ui, a.w);
            }
            *(float4*)(part + seg_col * 256 + c0) = a;
        }
        __syncthreads();
        if (t < N) {
            float s = 0.f;
#pragma unroll
            for (int sgi = 0; sgi < 16; ++sgi) s += part[sgi * 256 + t];
            v[t] = 1.0f / s;
        }
        __syncthreads();
    }

    // ---- Coalesced writeout: out[i][j] = u[i] * P0[i][j] * v[j] ----
#pragma unroll
    for (int k = 0; k < 16; ++k) {
        const int flat = k * 4096 + t * 4;
        const int r = flat >> 8;
        const int j = flat & 255;
        float4 p = *(const float4*)(P + r * STRIDE + j);
        float4 w = *(const float4*)(v + j);
        const float ur = u[r];                            // wave-broadcast
        p.x = p.x * w.x * ur;
        p.y = p.y * w.y * ur;
        p.z = p.z * w.z * ur;
        p.w = p.w * w.w * ur;
        *(float4*)(gout + flat) = p;
    }
}

extern "C" void kernel_launch(void* const* d_in, const int* in_sizes, int n_in,
                              void* d_out, int out_size, void* d_ws, size_t ws_size,
                              hipStream_t stream) {
    const float* in  = (const float*)d_in[0];
    float*       out = (float*)d_out;
    const int nmat = in_sizes[0] / (N * N);   // 2048

    (void)d_ws; (void)ws_size; (void)n_in; (void)out_size;

    hipFuncSetAttribute((const void*)sinkhorn_kernel,
                        hipFuncAttributeMaxDynamicSharedMemorySize, LDS_BYTES);
    sinkhorn_kernel<<<nmat, 1024, LDS_BYTES, stream>>>(in, out);
}